// TinyGraphClassifier_6133213298713
// MI455X (gfx1250) — compile-verified
//
#include <hip/hip_runtime.h>

typedef __attribute__((ext_vector_type(2))) float v2f;
typedef __attribute__((ext_vector_type(8))) float v8f;

namespace {
constexpr int kBlock1 = 256;
constexpr int kNPT    = 8;   // contiguous nodes per lane: 8 * 16B = one 128B cacheline of x
}

// Phase 1: streaming segment-sum of raw x + counts, run-length compressed atomics.
__global__ void __launch_bounds__(kBlock1) seg_pool_kernel(
    const float4* __restrict__ x, const int* __restrict__ gi,
    float* __restrict__ seg, float* __restrict__ cnt, int n) {
  const size_t start = ((size_t)blockIdx.x * kBlock1 + threadIdx.x) * (size_t)kNPT;
  if (start >= (size_t)n) return;

  int g_run = -1;
  float ax = 0.f, ay = 0.f, az = 0.f, aw = 0.f, ac = 0.f;

#pragma unroll
  for (int i = 0; i < kNPT; ++i) {
    size_t node = start + (size_t)i;
    if (node >= (size_t)n) break;
    int    g = gi[node];
    float4 v = x[node];
    if (g != g_run) {
      if (g_run >= 0) {
        atomicAdd(&seg[(size_t)g_run * 4 + 0], ax);
        atomicAdd(&seg[(size_t)g_run * 4 + 1], ay);
        atomicAdd(&seg[(size_t)g_run * 4 + 2], az);
        atomicAdd(&seg[(size_t)g_run * 4 + 3], aw);
        atomicAdd(&cnt[g_run], ac);
      }
      g_run = g; ax = v.x; ay = v.y; az = v.z; aw = v.w; ac = 1.f;
    } else {
      ax += v.x; ay += v.y; az += v.z; aw += v.w; ac += 1.f;
    }
  }
  if (g_run >= 0) {
    atomicAdd(&seg[(size_t)g_run * 4 + 0], ax);
    atomicAdd(&seg[(size_t)g_run * 4 + 1], ay);
    atomicAdd(&seg[(size_t)g_run * 4 + 2], az);
    atomicAdd(&seg[(size_t)g_run * 4 + 3], aw);
    atomicAdd(&cnt[g_run], ac);
  }
}

// Phase 2: out[g] = mean_x[g] @ (W_enc@W_head) + (b_enc@W_head + b_head)
// via V_WMMA_F32_16X16X4_F32 — one wave handles 16 graphs.
__global__ void __launch_bounds__(256) head_wmma_kernel(
    const float* __restrict__ seg, const float* __restrict__ cnt,
    const float* __restrict__ W_enc, const float* __restrict__ b_enc,
    const float* __restrict__ W_head, const float* __restrict__ b_head,
    float* __restrict__ out, int G) {
  // Fold the two linear layers: Wc[k][j2] = sum_j W_enc[k,j] * W_head[j,j2]
  float w00 = 0.f, w01 = 0.f, w10 = 0.f, w11 = 0.f;
  float w20 = 0.f, w21 = 0.f, w30 = 0.f, w31 = 0.f;
  float bc0 = b_head[0], bc1 = b_head[1];
#pragma unroll
  for (int j = 0; j < 4; ++j) {
    float h0 = W_head[j * 2 + 0], h1 = W_head[j * 2 + 1];
    w00 += W_enc[0 * 4 + j] * h0;  w01 += W_enc[0 * 4 + j] * h1;
    w10 += W_enc[1 * 4 + j] * h0;  w11 += W_enc[1 * 4 + j] * h1;
    w20 += W_enc[2 * 4 + j] * h0;  w21 += W_enc[2 * 4 + j] * h1;
    w30 += W_enc[3 * 4 + j] * h0;  w31 += W_enc[3 * 4 + j] * h1;
    bc0 += b_enc[j] * h0;          bc1 += b_enc[j] * h1;
  }

  const int  lane  = threadIdx.x & 31;
  const int  wave  = threadIdx.x >> 5;
  const int  gbase = (blockIdx.x * (int)(blockDim.x >> 5) + wave) * 16;

  const int  m  = lane & 15;      // A: row M; B/D: column N
  const bool hi = lane >= 16;
  const int  kb = hi ? 2 : 0;     // K pair held by this lane half (ISA 16x4 f32 A layout)

  // A matrix: 16 pooled means (16x4 f32). v0 holds K=kb, v1 holds K=kb+1.
  int   g   = gbase + m;
  int   gc  = (g < G) ? g : (G - 1);
  float inv = 1.0f / fmaxf(cnt[gc], 1.0f);
  float a0  = seg[(size_t)gc * 4 + kb + 0] * inv;
  float a1  = seg[(size_t)gc * 4 + kb + 1] * inv;
  if (g >= G) { a0 = 0.f; a1 = 0.f; }
  v2f A; A.x = a0; A.y = a1;

  // B matrix: 4x16, rows = Wc, columns >= 2 zero. v0 row K=kb, v1 row K=kb+1.
  float bx = 0.f, by = 0.f;
  if (m == 0)      { bx = hi ? w20 : w00;  by = hi ? w30 : w10; }
  else if (m == 1) { bx = hi ? w21 : w01;  by = hi ? w31 : w11; }
  v2f B; B.x = bx; B.y = by;

  v8f C = {};
  v8f D = __builtin_amdgcn_wmma_f32_16x16x4_f32(
      /*neg_a=*/false, A, /*neg_b=*/false, B,
      /*c_mod=*/(short)0, C, /*reuse_a=*/false, /*reuse_b=*/false);

  // D layout: VGPR j, lanes 0-15 -> M=j, lanes 16-31 -> M=j+8, N=lane&15.
  // Lanes with N<2 hold a full output column across the 8 accumulator VGPRs.
  if (m < 2) {
    float bcv = (m == 0) ? bc0 : bc1;
    int   mb  = hi ? 8 : 0;
#pragma unroll
    for (int j = 0; j < 8; ++j) {
      int go = gbase + mb + j;
      if (go < G) out[(size_t)go * 2 + m] = D[j] + bcv;
    }
  }
}

extern "C" void kernel_launch(void* const* d_in, const int* in_sizes, int n_in,
                              void* d_out, int out_size, void* d_ws, size_t ws_size,
                              hipStream_t stream) {
  const float* x      = (const float*)d_in[0];
  const int*   gi     = (const int*)d_in[1];
  const float* W_enc  = (const float*)d_in[2];
  const float* b_enc  = (const float*)d_in[3];
  const float* W_head = (const float*)d_in[4];
  const float* b_head = (const float*)d_in[5];
  float*       out    = (float*)d_out;

  const int n = in_sizes[1];      // number of nodes
  const int G = out_size / 2;     // number of graphs

  float* seg = (float*)d_ws;                 // [G][4] sums
  float* cnt = seg + (size_t)G * 4;          // [G] counts
  hipMemsetAsync(d_ws, 0, (size_t)G * 5 * sizeof(float), stream);

  const int nodes_per_block = kBlock1 * kNPT;
  const int grid1 = (n + nodes_per_block - 1) / nodes_per_block;
  seg_pool_kernel<<<grid1, kBlock1, 0, stream>>>(
      (const float4*)x, gi, seg, cnt, n);

  const int waves = (G + 15) / 16;
  const int grid2 = (waves + 7) / 8;         // 8 waves (256 threads) per block
  head_wmma_kernel<<<grid2, 256, 0, stream>>>(
      seg, cnt, W_enc, b_enc, W_head, b_head, out, G);
}